// AnnealedSMC_20237885899158
// MI455X (gfx1250) — compile-verified
//
#include <hip/hip_runtime.h>
#include <math.h>

#define DD 64
#define TT 8
#define KK 10

typedef __attribute__((ext_vector_type(2))) float v2f;
typedef __attribute__((ext_vector_type(8))) float v8f;

// ---------------- stateless deterministic RNG ----------------
__device__ __forceinline__ unsigned hmix(unsigned x) {
  x ^= x >> 16; x *= 0x7feb352dU; x ^= x >> 15; x *= 0x846ca68bU; x ^= x >> 16;
  return x;
}
__device__ __forceinline__ unsigned hash4(unsigned a, unsigned b, unsigned c, unsigned d) {
  unsigned h = a * 0x9E3779B9u ^ b * 0x85EBCA6Bu ^ c * 0xC2B2AE35u ^ d * 0x27D4EB2Fu;
  return hmix(h ^ (a + b + c + d));
}
__device__ __forceinline__ float u01f(unsigned h) {
  return ((float)(h >> 8) + 0.5f) * 5.9604644775390625e-08f;  // (0,1)
}
__device__ __forceinline__ float nrmf(unsigned h1, unsigned h2) {
  float u1 = u01f(h1), u2 = u01f(h2);
  return sqrtf(-2.0f * logf(u1)) * cosf(6.28318530717958647f * u2);
}

// ---------------- async staging of means into LDS (gfx1250 async-to-LDS path) ----
// s_means[0..63] = init_mean, s_means[64..127] = target_mean (512 bytes total).
// Wave 0: 32 lanes x b128 = 512 bytes in a single ASYNCcnt-tracked instruction.
__device__ __forceinline__ void stage_means_async(float* s_means,
                                                  const float* __restrict__ im,
                                                  const float* __restrict__ tm) {
  if (threadIdx.x < 32) {
    int lane = threadIdx.x;
    const float* src = (lane < 16) ? (im + lane * 4) : (tm + (lane - 16) * 4);
    unsigned ldsoff = (unsigned)(size_t)s_means + (unsigned)lane * 16u;  // LDS byte address of this lane's 16 bytes
    unsigned long long ga = (unsigned long long)(size_t)src;
    asm volatile("global_load_async_to_lds_b128 %0, %1, off"
                 :: "v"(ldsoff), "v"(ga) : "memory");
    asm volatile("s_wait_asynccnt 0" ::: "memory");
  }
  __syncthreads();
}

// ---------------- init: x0 = init_mean + N(0,I), w0 = 1/N ----------------
__global__ __launch_bounds__(256) void smc_init_kernel(
    float* __restrict__ x0, float* __restrict__ w0,
    const float* __restrict__ init_mean, int N) {
  int i = blockIdx.x * 256 + threadIdx.x;
  if (i < N * DD) {
    int d = i & (DD - 1);
    int p = i >> 6;
    unsigned h1 = hash4(0xA11CE5u, 0u, (unsigned)p, (unsigned)(2 * d));
    unsigned h2 = hash4(0xA11CE5u, 0u, (unsigned)p, (unsigned)(2 * d + 1));
    x0[i] = init_mean[d] + nrmf(h1, h2);
  }
  if (i < N) w0[i] = 1.0f / (float)N;
}

// ---------------- 10 MALA steps at temperature t, state held in VGPRs ----------------
__global__ __launch_bounds__(256) void smc_propagate_kernel(
    const float* __restrict__ xin, float* __restrict__ xout,
    const float* __restrict__ init_mean, const float* __restrict__ target_mean,
    int N, int t) {
  __shared__ float s_means[2 * DD];
  stage_means_async(s_means, init_mean, target_mean);
  const float* s_im = s_means;
  const float* s_tm = s_means + DD;

  int p = blockIdx.x * 256 + threadIdx.x;
  if (p >= N) return;

  float x[DD];
#pragma unroll
  for (int d = 0; d < DD; ++d) x[d] = xin[(size_t)p * DD + d];

  const float tf = (float)t;
  const float wT = -2.0f * tf;              // t * d/dx tgt_lp = -t*(x-tm)/0.5
  const float wI = -((float)TT - tf);       // (T-t) * d/dx init_lp

#pragma unroll 1
  for (int k = 0; k < KK; ++k) {
    float tau = 1.0f / ((float)(k + 1) * (float)DD);
    float s2t = sqrtf(2.0f * tau);
    float nx[DD];
    float at_x = 0.f, ai_x = 0.f, at_n = 0.f, ai_n = 0.f, rrev = 0.f, rfwd = 0.f;
#pragma unroll
    for (int d = 0; d < DD; ++d) {
      float tm = s_tm[d], im = s_im[d];
      float dxT = x[d] - tm, dxI = x[d] - im;
      float g = wT * dxT + wI * dxI;
      float mfwd = x[d] + tau * g;
      unsigned h1 = hash4(0x1000u + (unsigned)t, (unsigned)k, (unsigned)p, (unsigned)(2 * d));
      unsigned h2 = hash4(0x1000u + (unsigned)t, (unsigned)k, (unsigned)p, (unsigned)(2 * d + 1));
      float n = nrmf(h1, h2);
      float xn = mfwd + s2t * n;
      nx[d] = xn;
      float dnT = xn - tm, dnI = xn - im;
      float gn = wT * dnT + wI * dnI;
      float mrev = xn + tau * gn;
      at_x = fmaf(dxT, dxT, at_x);
      ai_x = fmaf(dxI, dxI, ai_x);
      at_n = fmaf(dnT, dnT, at_n);
      ai_n = fmaf(dnI, dnI, ai_n);
      float dr = x[d] - mrev;  rrev = fmaf(dr, dr, rrev);
      float df = xn - mfwd;    rfwd = fmaf(df, df, rfwd);
    }
    // L(new)-L(x):  -0.5*t*(dAt)/sigma2 - 0.5*(T-t)*dAi   (sigma2=0.5)
    float dL = -tf * (at_n - at_x) - 0.5f * ((float)TT - tf) * (ai_n - ai_x);
    float log_alpha = dL + (rfwd - rrev) * (0.25f / tau);
    unsigned hu = hash4(0xBEEFu + (unsigned)t, (unsigned)k, (unsigned)p, 0xFFFFu);
    bool accept = logf(u01f(hu)) < log_alpha;
#pragma unroll
    for (int d = 0; d < DD; ++d) x[d] = accept ? nx[d] : x[d];
  }
#pragma unroll
  for (int d = 0; d < DD; ++d) xout[(size_t)p * DD + d] = x[d];
}

// ---------------- inclusive scan over weights (3 phases) ----------------
__global__ __launch_bounds__(256) void smc_scan1_kernel(
    const float* __restrict__ w, float* __restrict__ cdf,
    float* __restrict__ bsum, int N) {
  __shared__ float tmp[256];
  int i = blockIdx.x * 256 + threadIdx.x;
  float v = (i < N) ? w[i] : 0.f;
  tmp[threadIdx.x] = v;
  __syncthreads();
  for (int off = 1; off < 256; off <<= 1) {
    float add = (threadIdx.x >= off) ? tmp[threadIdx.x - off] : 0.f;
    __syncthreads();
    tmp[threadIdx.x] += add;
    __syncthreads();
  }
  if (i < N) cdf[i] = tmp[threadIdx.x];
  if (threadIdx.x == 255) bsum[blockIdx.x] = tmp[255];
}

__global__ __launch_bounds__(256) void smc_scan2_kernel(
    const float* __restrict__ bsum, float* __restrict__ boff, int nb) {
  __shared__ float tmp[256];
  float carry = 0.f;
  for (int base = 0; base < nb; base += 256) {
    int i = base + threadIdx.x;
    float v = (i < nb) ? bsum[i] : 0.f;
    tmp[threadIdx.x] = v;
    __syncthreads();
    for (int off = 1; off < 256; off <<= 1) {
      float add = (threadIdx.x >= off) ? tmp[threadIdx.x - off] : 0.f;
      __syncthreads();
      tmp[threadIdx.x] += add;
      __syncthreads();
    }
    if (i < nb) boff[i] = carry + tmp[threadIdx.x] - v;  // exclusive
    __syncthreads();
    carry += tmp[255];
    __syncthreads();
  }
}

__global__ __launch_bounds__(256) void smc_scan3_kernel(
    float* __restrict__ cdf, const float* __restrict__ boff, int N) {
  int i = blockIdx.x * 256 + threadIdx.x;
  if (i < N) cdf[i] += boff[blockIdx.x];
}

// ---------------- multinomial resample: binary-search CDF + gather ----------------
__global__ __launch_bounds__(256) void smc_resample_kernel(
    const float* __restrict__ cdf, const float* __restrict__ xp,
    float* __restrict__ xr, int N, int t) {
  int i = blockIdx.x * 256 + threadIdx.x;
  if (i >= N) return;
  float total = cdf[N - 1];
  unsigned h = hash4(0x5EED5u + (unsigned)t, 7777u, (unsigned)i, 3u);
  float u = u01f(h) * total;
  int lo = 0, hi = N - 1;
  while (lo < hi) {
    int mid = (lo + hi) >> 1;
    if (cdf[mid] < u) lo = mid + 1; else hi = mid;
  }
  const float* src = xp + (size_t)lo * DD;
  float* dst = xr + (size_t)i * DD;
#pragma unroll
  for (int d = 0; d < DD; ++d) dst[d] = src[d];
}

// ---------------- reweight logits via fp32 WMMA row-sum reduction ----------------
// Each wave32 handles 16 particles. S_t = sum_d (x-tm)^2 and S_i = sum_d (x-im)^2
// as 16 chunks of [16x4 f32] x [4x16 ones] V_WMMA_F32_16X16X4_F32, f32 exact.
// A layout (ISA 7.12.2, 32-bit A 16x4): lanes 0-15 M=0..15 hold K={0,1}; lanes 16-31 K={2,3}.
// ulw = -S_t + 0.5*S_i + 0.5*D*ln(2) - ln(N)   (sigma2 = 0.5)
__global__ __launch_bounds__(256) void smc_ulw_wmma_kernel(
    const float* __restrict__ xr,
    const float* __restrict__ init_mean, const float* __restrict__ target_mean,
    float* __restrict__ ulw, float* __restrict__ blockmax, int N) {
  __shared__ float s_means[2 * DD];
  __shared__ float smax[16];
  stage_means_async(s_means, init_mean, target_mean);
  const float* s_im = s_means;
  const float* s_tm = s_means + DD;

  int lane = threadIdx.x & 31;
  int wave = threadIdx.x >> 5;
  int pbase = blockIdx.x * 128 + wave * 16;
  int M = lane & 15;
  int hi = lane >> 4;
  int p = pbase + M;
  int pc = (p < N) ? p : (N - 1);  // clamp: keep EXEC full for WMMA
  const float* xrow = xr + (size_t)pc * DD;

  v2f bones; bones[0] = 1.0f; bones[1] = 1.0f;
  v8f c_t = {}; v8f c_i = {};
#pragma unroll
  for (int c = 0; c < 16; ++c) {
    int k = 4 * c + 2 * hi;
    float x0 = xrow[k], x1 = xrow[k + 1];
    float dt0 = x0 - s_tm[k], dt1 = x1 - s_tm[k + 1];
    float di0 = x0 - s_im[k], di1 = x1 - s_im[k + 1];
    v2f at; at[0] = dt0 * dt0; at[1] = dt1 * dt1;
    v2f ai; ai[0] = di0 * di0; ai[1] = di1 * di1;
    c_t = __builtin_amdgcn_wmma_f32_16x16x4_f32(false, at, false, bones, (short)0, c_t, false, false);
    c_i = __builtin_amdgcn_wmma_f32_16x16x4_f32(false, ai, false, bones, (short)0, c_i, false, false);
  }

  // C/D layout: VGPR r, lanes 0-15 -> M=r ; lanes 16-31 -> M=8+r. Lane 0/16 hold N=0.
  const float cadd = 0.5f * (float)DD * 0.69314718056f - logf((float)N);
  if (lane == 0 || lane == 16) {
    int baseM = pbase + hi * 8;
    float lm = -3.4e38f;
#pragma unroll
    for (int r = 0; r < 8; ++r) {
      int pi = baseM + r;
      if (pi < N) {
        float v = -c_t[r] + 0.5f * c_i[r] + cadd;
        ulw[pi] = v;
        lm = fmaxf(lm, v);
      }
    }
    smax[wave * 2 + hi] = lm;
  }
  __syncthreads();
  if (threadIdx.x == 0) {
    float m = smax[0];
#pragma unroll
    for (int j = 1; j < 16; ++j) m = fmaxf(m, smax[j]);
    blockmax[blockIdx.x] = m;
  }
}

// ---------------- softmax pieces ----------------
__global__ __launch_bounds__(256) void smc_reduce_max_kernel(
    const float* __restrict__ in, int n, float* __restrict__ out) {
  __shared__ float tmp[256];
  float m = -3.4e38f;
  for (int i = threadIdx.x; i < n; i += 256) m = fmaxf(m, in[i]);
  tmp[threadIdx.x] = m;
  __syncthreads();
  for (int off = 128; off > 0; off >>= 1) {
    if (threadIdx.x < off) tmp[threadIdx.x] = fmaxf(tmp[threadIdx.x], tmp[threadIdx.x + off]);
    __syncthreads();
  }
  if (threadIdx.x == 0) out[0] = tmp[0];
}

__global__ __launch_bounds__(256) void smc_exp_blocksum_kernel(
    const float* __restrict__ ulw, const float* __restrict__ gmax,
    float* __restrict__ wout, float* __restrict__ bsum, int N) {
  __shared__ float tmp[256];
  int i = blockIdx.x * 256 + threadIdx.x;
  float e = 0.f;
  if (i < N) {
    e = expf(ulw[i] - gmax[0]);
    wout[i] = e;
  }
  tmp[threadIdx.x] = e;
  __syncthreads();
  for (int off = 128; off > 0; off >>= 1) {
    if (threadIdx.x < off) tmp[threadIdx.x] += tmp[threadIdx.x + off];
    __syncthreads();
  }
  if (threadIdx.x == 0) bsum[blockIdx.x] = tmp[0];
}

__global__ __launch_bounds__(256) void smc_reduce_sum_inv_kernel(
    const float* __restrict__ in, int n, float* __restrict__ out) {
  __shared__ float tmp[256];
  float s = 0.f;
  for (int i = threadIdx.x; i < n; i += 256) s += in[i];
  tmp[threadIdx.x] = s;
  __syncthreads();
  for (int off = 128; off > 0; off >>= 1) {
    if (threadIdx.x < off) tmp[threadIdx.x] += tmp[threadIdx.x + off];
    __syncthreads();
  }
  if (threadIdx.x == 0) out[0] = 1.0f / tmp[0];
}

__global__ __launch_bounds__(256) void smc_normalize_kernel(
    float* __restrict__ w, const float* __restrict__ inv, int N) {
  int i = blockIdx.x * 256 + threadIdx.x;
  if (i < N) w[i] *= inv[0];
}

// ---------------- host orchestration ----------------
extern "C" void kernel_launch(void* const* d_in, const int* in_sizes, int n_in,
                              void* d_out, int out_size, void* d_ws, size_t ws_size,
                              hipStream_t stream) {
  const float* im = (const float*)d_in[0];
  const float* tm = (const float*)d_in[1];
  (void)in_sizes; (void)n_in; (void)ws_size;

  int N = out_size / (TT * (DD + 1));  // out = [T,N,D] particles + [T,N] weights
  float* particles = (float*)d_out;
  float* weights = particles + (size_t)TT * N * DD;

  int nb256 = (N + 255) / 256;
  int nbU = (N + 127) / 128;
  int nbred = (nbU > nb256) ? nbU : nb256;

  float* wsf = (float*)d_ws;
  size_t off = 0;
  float* xp   = wsf + off; off += (size_t)N * DD;
  float* ulw  = wsf + off; off += (size_t)N;
  float* cdf  = wsf + off; off += (size_t)N;
  float* bsum = wsf + off; off += (size_t)nb256;
  float* boff = wsf + off; off += (size_t)nb256;
  float* bred = wsf + off; off += (size_t)nbred;
  float* sc   = wsf + off; off += 2;

  smc_init_kernel<<<(N * DD + 255) / 256, 256, 0, stream>>>(particles, weights, im, N);

  for (int t = 1; t < TT; ++t) {
    float* xprev = particles + (size_t)(t - 1) * N * DD;
    float* xcur  = particles + (size_t)t * N * DD;
    float* wprev = weights + (size_t)(t - 1) * N;
    float* wcur  = weights + (size_t)t * N;

    smc_propagate_kernel<<<(N + 255) / 256, 256, 0, stream>>>(xprev, xp, im, tm, N, t);
    smc_scan1_kernel<<<nb256, 256, 0, stream>>>(wprev, cdf, bsum, N);
    smc_scan2_kernel<<<1, 256, 0, stream>>>(bsum, boff, nb256);
    smc_scan3_kernel<<<nb256, 256, 0, stream>>>(cdf, boff, N);
    smc_resample_kernel<<<(N + 255) / 256, 256, 0, stream>>>(cdf, xp, xcur, N, t);
    smc_ulw_wmma_kernel<<<nbU, 256, 0, stream>>>(xcur, im, tm, ulw, bred, N);
    smc_reduce_max_kernel<<<1, 256, 0, stream>>>(bred, nbU, sc);
    smc_exp_blocksum_kernel<<<nb256, 256, 0, stream>>>(ulw, sc, wcur, bred, N);
    smc_reduce_sum_inv_kernel<<<1, 256, 0, stream>>>(bred, nb256, sc + 1);
    smc_normalize_kernel<<<nb256, 256, 0, stream>>>(wcur, sc + 1, N);
  }
}